// ProceduralMemory_82325933130216
// MI455X (gfx1250) — compile-verified
//
#include <hip/hip_runtime.h>
#include <math.h>

typedef __attribute__((ext_vector_type(2))) float v2f;
typedef __attribute__((ext_vector_type(8))) float v8f;
typedef __attribute__((ext_vector_type(4))) unsigned int u32x4;
typedef __attribute__((ext_vector_type(8))) unsigned int u32x8;

#define D_PM   64
#define DDIM   1024
#define NSEQ   4096
#define BSZ    8
#define BMAT   8
#define OUT_W  (BSZ*NSEQ*DDIM)

// ws layout (float offsets)
#define WS_G    0        // 8*64*64      = 32768
#define WS_WSUM 32768    // 8*64*64      = 32768
#define WS_PIT  65536    // 1024*64      = 65536
#define WS_CT   131072   // 8*64*1024    = 524288
// total 655360 floats = 2.62 MB scratch

__device__ __forceinline__ v8f wmma_f32(v2f a, v2f b, v8f c) {
  // D = A(16x4, f32) * B(4x16, f32) + C(16x16, f32)
  return __builtin_amdgcn_wmma_f32_16x16x4_f32(false, a, false, b, (short)0, c,
                                               false, false);
}

// ---- k0a: zero G, W_sum = sum_k W_pm, piT = proj_in_w^T --------------------
__global__ __launch_bounds__(256) void k0a(const float* __restrict__ wpm,
                                           const float* __restrict__ pinw,
                                           float* __restrict__ ws) {
  int i = blockIdx.x * 256 + threadIdx.x;        // 65536 threads
  if (i < 32768) {
    ws[WS_G + i] = 0.f;
    int b = i >> 12, r = i & 4095;
    float s = 0.f;
#pragma unroll
    for (int k = 0; k < BMAT; ++k) s += wpm[b * 32768 + k * 4096 + r];
    ws[WS_WSUM + i] = s;
  }
  {
    int d = i >> 6, p = i & 63;
    ws[WS_PIT + i] = pinw[p * DDIM + d];
  }
}

// ---- k0b: Ct[b][p][d] = sum_q Wsum[b][q][p] * proj_out_w[d][q] -------------
__global__ __launch_bounds__(256) void k0b(const float* __restrict__ poutw,
                                           float* __restrict__ ws) {
  int i = blockIdx.x * 256 + threadIdx.x;        // 524288 threads
  int b = i >> 16, p = (i >> 10) & 63, d = i & 1023;
  const float* wsum = ws + WS_WSUM + b * 4096;
  float s = 0.f;
#pragma unroll
  for (int q = 0; q < 64; ++q) s += wsum[q * 64 + p] * poutw[d * 64 + q];
  ws[WS_CT + i] = s;
}

// ---- k1: fused streaming kernel (pre, surprise gate, G, pm_read) -----------
// smem partition (floats): Hs 16*1028 (TDM-padded) | pres 16*68 | wts 16*68
//                          | rowpart 128 | roww 16
#define SM_PRES 16448
#define SM_WTS  17536
#define SM_RP   18624
#define SM_RW   18752
#define K1_SMEM (18768 * 4)

// TDM pads 1 dword after every 256 dwords -> element (row,col) lives at:
#define HIDX(row, col) ((row) * 1028 + (col) + ((col) >> 8))

__global__ __launch_bounds__(256) void k1(const float* __restrict__ H,
                                          const float* __restrict__ surp,
                                          const float* __restrict__ pinb,
                                          const float* __restrict__ poutb,
                                          const float* __restrict__ sscale,
                                          const float* __restrict__ sbias,
                                          float* __restrict__ ws,
                                          float* __restrict__ out) {
  extern __shared__ float smem[];
  float* Hs = smem;
  float* pres = smem + SM_PRES;
  float* wts = smem + SM_WTS;
  float* rowpart = smem + SM_RP;
  float* roww = smem + SM_RW;

  const int tid = threadIdx.x;
  const int wave = tid >> 5, lane = tid & 31;
  const int hf = lane >> 4, ln = lane & 15;
  const int b = blockIdx.y;
  const int n0 = blockIdx.x * 16;

  const float* piT = ws + WS_PIT;
  const float* Ct = ws + WS_CT + (size_t)b * 64 * DDIM;
  float* G = ws + WS_G + b * 4096;

  // ---- stage contiguous 16x1024 f32 H tile into LDS via Tensor Data Mover.
  // One TDM op per workgroup (wave 0, uniform branch). D# (ISA 8.3/8.4):
  //  group0: count=1 | lds_addr=0 | global_addr | type=2
  //  group1: data_size=4B, pad_enable, pad_interval=256dw, pad_amount=1dw,
  //          tensor_dim0=1024, tensor_dim1=16, tile_dim0=1024, tile_dim1=16,
  //          tensor_dim0_stride=1024
  if (wave == 0) {
    unsigned long long ga =
        (unsigned long long)(H + (size_t)(b * NSEQ + n0) * DDIM);
    u32x4 g0;
    g0[0] = 1u;                                        // count=1, no gather
    g0[1] = 0u;                                        // lds_addr (bytes)
    g0[2] = (unsigned int)ga;                          // global_addr[31:0]
    g0[3] = (unsigned int)(ga >> 32) | 0x80000000u;    // addr[56:32] | type=2
    u32x8 g1;
    g1[0] = (2u << 16) | (1u << 20) | (7u << 22);      // 4B, pad 1dw/256dw
    g1[1] = (DDIM & 0xffffu) << 16;                    // tensor_dim0 lo16
    g1[2] = 16u << 16;                                 // tensor_dim1 = 16 rows
    g1[3] = (DDIM & 0xffffu) << 16;                    // tile_dim0 = 1024
    g1[4] = 16u;                                       // tile_dim1 = 16
    g1[5] = DDIM;                                      // dim0_stride = 1024
    g1[6] = 0u;
    g1[7] = 0u;
    asm volatile("tensor_load_to_lds %0, %1" ::"s"(g0), "s"(g1) : "memory");
    __builtin_amdgcn_s_wait_tensorcnt(0);
  }
  __syncthreads();

  if (wave < 4) {
    // waves 0-3: pre[n,p] = H[n,:] . proj_in_w[p,:]  (WMMA, K=1024, 2 accums)
    const int p0 = wave * 16;
    v8f acc0 = {}; v8f acc1 = {};
    for (int k = 0; k < DDIM; k += 8) {
      int ka = k + 2 * hf;
      v2f a0, b0;
      a0.x = Hs[HIDX(ln, ka)]; a0.y = Hs[HIDX(ln, ka) + 1];
      b0.x = piT[ka * 64 + p0 + ln]; b0.y = piT[(ka + 1) * 64 + p0 + ln];
      acc0 = wmma_f32(a0, b0, acc0);
      int kb = k + 4 + 2 * hf;
      v2f a1, b1;
      a1.x = Hs[HIDX(ln, kb)]; a1.y = Hs[HIDX(ln, kb) + 1];
      b1.x = piT[kb * 64 + p0 + ln]; b1.y = piT[(kb + 1) * 64 + p0 + ln];
      acc1 = wmma_f32(a1, b1, acc1);
    }
    v8f acc = acc0 + acc1;
    float bias = pinb[p0 + ln];
#pragma unroll
    for (int r = 0; r < 8; ++r)
      pres[(r + 8 * hf) * 68 + p0 + ln] = acc[r] + bias;
  } else {
    // waves 4-7: surprise row sum-of-squares (8 threads per row)
    int t = tid - 128;
    int row = t >> 3, j = t & 7;
    const float4* sp =
        (const float4*)(surp + (size_t)(b * NSEQ + n0 + row) * DDIM);
    float s = 0.f;
#pragma unroll 4
    for (int i = 0; i < 32; ++i) {
      float4 v = sp[j + 8 * i];
      s += v.x * v.x + v.y * v.y + v.z * v.z + v.w * v.w;
    }
    rowpart[row * 8 + j] = s;
  }
  __syncthreads();

  if (tid < 16) {
    float s = 0.f;
#pragma unroll
    for (int j = 0; j < 8; ++j) s += rowpart[tid * 8 + j];
    float mag = sqrtf(s);
    float z = sscale[0] * mag + sbias[0];
    roww[tid] = sqrtf(1.f / (1.f + expf(-z)));
  }
  __syncthreads();

  // weighted = sqrt(s) * pre
#pragma unroll
  for (int i = 0; i < 4; ++i) {
    int e = tid * 4 + i;
    int row = e >> 6, p = e & 63;
    wts[row * 68 + p] = roww[row] * pres[row * 68 + p];
  }
  __syncthreads();

  // G partial: wts^T @ wts (64x64, K=16), 16 tiles over 8 waves, atomic add
#pragma unroll
  for (int i = 0; i < 2; ++i) {
    int t = wave * 2 + i;
    int p0g = (t >> 2) * 16, q0g = (t & 3) * 16;
    v8f acc = {};
#pragma unroll
    for (int kk = 0; kk < 4; ++kk) {
      int k0 = kk * 4 + 2 * hf;
      v2f a, bb;
      a.x = wts[k0 * 68 + p0g + ln]; a.y = wts[(k0 + 1) * 68 + p0g + ln];
      bb.x = wts[k0 * 68 + q0g + ln]; bb.y = wts[(k0 + 1) * 68 + q0g + ln];
      acc = wmma_f32(a, bb, acc);
    }
#pragma unroll
    for (int r = 0; r < 8; ++r)
      atomicAdd(&G[(p0g + r + 8 * hf) * 64 + q0g + ln], acc[r]);
  }

  // pm_read = pre @ Ct + proj_out_b : 64 output tiles (16 d-cols each)
  for (int t = 0; t < 8; ++t) {
    int d0 = (wave * 8 + t) * 16;
    v8f acc = {};
    for (int kk = 0; kk < 16; ++kk) {
      int k0 = kk * 4 + 2 * hf;
      v2f a, bb;
      a.x = pres[ln * 68 + k0]; a.y = pres[ln * 68 + k0 + 1];
      bb.x = Ct[(size_t)k0 * DDIM + d0 + ln];
      bb.y = Ct[(size_t)(k0 + 1) * DDIM + d0 + ln];
      acc = wmma_f32(a, bb, acc);
    }
    float ob = poutb[d0 + ln];
    float* orow = out + (size_t)(b * NSEQ + n0) * DDIM + d0 + ln;
#pragma unroll
    for (int r = 0; r < 8; ++r)
      orow[(size_t)(r + 8 * hf) * DDIM] = acc[r] + ob;
  }
}

// ---- k2: W_new = W_pm @ (decay*I + beta*G/N), Frobenius budget clamp -------
#define K2_SMEM (8705 * 4)
__global__ __launch_bounds__(256) void k2(const float* __restrict__ wpm,
                                          const float* __restrict__ rbeta,
                                          const float* __restrict__ rdecay,
                                          const float* __restrict__ ws,
                                          float* __restrict__ out) {
  extern __shared__ float smem[];
  float* Ws = smem;            // 64*68
  float* Ts = smem + 4352;     // 64*68
  float* fr = smem + 8704;     // 1
  int tid = threadIdx.x;
  int wave = tid >> 5, lane = tid & 31, hf = lane >> 4, ln = lane & 15;
  int blk = blockIdx.x;        // b*8 + k
  int b = blk >> 3, kslot = blk & 7;
  float beta = log1pf(expf(rbeta[kslot]));
  float decay = 1.f / (1.f + expf(-rdecay[kslot]));
  const float* wsrc = wpm + (size_t)blk * 4096;
  const float* g = ws + WS_G + b * 4096;
#pragma unroll
  for (int i = 0; i < 16; ++i) {
    int e = tid + i * 256;
    int r = e >> 6, c = e & 63;
    Ws[r * 68 + c] = wsrc[e];
    Ts[r * 68 + c] = (r == c ? decay : 0.f) + beta * (g[e] * (1.f / (float)NSEQ));
  }
  if (tid == 0) fr[0] = 0.f;
  __syncthreads();

  v8f accs[2];
  float ss = 0.f;
#pragma unroll
  for (int i = 0; i < 2; ++i) {
    int t = wave * 2 + i;
    int m0 = (t >> 2) * 16, q0 = (t & 3) * 16;
    v8f acc = {};
#pragma unroll
    for (int kk = 0; kk < 16; ++kk) {
      int k0 = kk * 4 + 2 * hf;
      v2f a, bb;
      a.x = Ws[(m0 + ln) * 68 + k0]; a.y = Ws[(m0 + ln) * 68 + k0 + 1];
      bb.x = Ts[k0 * 68 + q0 + ln];  bb.y = Ts[(k0 + 1) * 68 + q0 + ln];
      acc = wmma_f32(a, bb, acc);
    }
    accs[i] = acc;
#pragma unroll
    for (int r = 0; r < 8; ++r) ss += acc[r] * acc[r];
  }
  atomicAdd(fr, ss);
  __syncthreads();
  float sc = fminf(16.f / fmaxf(sqrtf(fr[0]), 1e-8f), 1.f);
  float* od = out + OUT_W + (size_t)blk * 4096;
#pragma unroll
  for (int i = 0; i < 2; ++i) {
    int t = wave * 2 + i;
    int m0 = (t >> 2) * 16, q0 = (t & 3) * 16;
#pragma unroll
    for (int r = 0; r < 8; ++r)
      od[(m0 + r + 8 * hf) * 64 + q0 + ln] = accs[i][r] * sc;
  }
}

extern "C" void kernel_launch(void* const* d_in, const int* in_sizes, int n_in,
                              void* d_out, int out_size, void* d_ws,
                              size_t ws_size, hipStream_t stream) {
  const float* H = (const float*)d_in[0];
  const float* surprise = (const float*)d_in[1];
  const float* W_pm = (const float*)d_in[2];
  const float* pin_w = (const float*)d_in[3];
  const float* pin_b = (const float*)d_in[4];
  const float* pout_w = (const float*)d_in[5];
  const float* pout_b = (const float*)d_in[6];
  const float* raw_beta = (const float*)d_in[7];
  const float* raw_decay = (const float*)d_in[8];
  const float* sscale = (const float*)d_in[9];
  const float* sbias = (const float*)d_in[10];
  float* out = (float*)d_out;
  float* ws = (float*)d_ws;

  k0a<<<256, 256, 0, stream>>>(W_pm, pin_w, ws);
  k0b<<<2048, 256, 0, stream>>>(pout_w, ws);
  dim3 g1(NSEQ / 16, BSZ);
  k1<<<g1, 256, K1_SMEM, stream>>>(H, surprise, pin_b, pout_b, sscale, sbias,
                                   ws, out);
  k2<<<64, 256, K2_SMEM, stream>>>(W_pm, raw_beta, raw_decay, ws, out);
}